// Attention_10883447128214
// MI455X (gfx1250) — compile-verified
//
#include <hip/hip_runtime.h>

#define BZ   32
#define SLEN 1024
#define TLEN 1024
#define DIM  512

typedef __attribute__((ext_vector_type(2))) float v2f;
typedef __attribute__((ext_vector_type(8))) float v8f;

// D = A(16x4) * B(4x16) + C, exact f32 (matches reference numerics)
static __device__ __forceinline__ v8f wmma4(v2f a, v2f b, v8f c) {
    return __builtin_amdgcn_wmma_f32_16x16x4_f32(false, a, false, b, (short)0, c,
                                                 false, false);
}

// Fragment load where source is row-major [row, k] with leading dim ld.
// Works for A (rows = M) and for B when B = S^T with S row-major [n, k].
// Lane l: r = l&15 selects row, h = l>>4 selects K-half; contiguous 8B load.
static __device__ __forceinline__ v2f frag_rk(const float* __restrict__ p,
                                              int row0, int ld, int k0, int lane) {
    const int r = lane & 15, h = lane >> 4;
    return *(const v2f*)(p + (size_t)(row0 + r) * ld + k0 + 2 * h);
}

// B fragment with B genuinely row-major [k, n] (strided: two b32 loads)
static __device__ __forceinline__ v2f frag_kn(const float* __restrict__ p,
                                              int k0, int n0, int ld, int lane) {
    const int n = lane & 15, h = lane >> 4;
    v2f b;
    b.x = p[(size_t)(k0 + 2 * h)     * ld + n0 + n];
    b.y = p[(size_t)(k0 + 2 * h + 1) * ld + n0 + n];
    return b;
}

// -------- Kernel 1: score = tgt @ src^T, masked, -> align region ----------
__global__ __launch_bounds__(256) void score_kernel(const float* __restrict__ src,
                                                    const float* __restrict__ tgt,
                                                    const int*  __restrict__ lens,
                                                    float* __restrict__ align) {
    const int lane = threadIdx.x & 31;
    int w = blockIdx.x * 8 + (threadIdx.x >> 5);
    const int nt = w & 15;  w >>= 4;   // 16 tiles of 64 along src
    const int mt = w & 31;  w >>= 5;   // 32 tiles of 32 along tgt
    const int b  = w;                  // batch
    const int t0 = mt * 32, s0 = nt * 64;
    const float* A = tgt + (size_t)b * TLEN * DIM;   // [t, d]
    const float* S = src + (size_t)b * SLEN * DIM;   // [s, d] -> B = S^T

    v8f acc[2][4] = {};
    #pragma unroll 4
    for (int k0 = 0; k0 < DIM; k0 += 4) {
        v2f a0 = frag_rk(A, t0,      DIM, k0, lane);
        v2f a1 = frag_rk(A, t0 + 16, DIM, k0, lane);
        v2f b0 = frag_rk(S, s0,      DIM, k0, lane);
        v2f b1 = frag_rk(S, s0 + 16, DIM, k0, lane);
        v2f b2 = frag_rk(S, s0 + 32, DIM, k0, lane);
        v2f b3 = frag_rk(S, s0 + 48, DIM, k0, lane);
        acc[0][0] = wmma4(a0, b0, acc[0][0]);
        acc[0][1] = wmma4(a0, b1, acc[0][1]);
        acc[0][2] = wmma4(a0, b2, acc[0][2]);
        acc[0][3] = wmma4(a0, b3, acc[0][3]);
        acc[1][0] = wmma4(a1, b0, acc[1][0]);
        acc[1][1] = wmma4(a1, b1, acc[1][1]);
        acc[1][2] = wmma4(a1, b2, acc[1][2]);
        acc[1][3] = wmma4(a1, b3, acc[1][3]);
    }
    const int len = lens[b];
    float* out = align + (size_t)b * TLEN * SLEN;
    const int n = lane & 15, h = lane >> 4;
    #pragma unroll
    for (int mi = 0; mi < 2; ++mi)
        #pragma unroll
        for (int ni = 0; ni < 4; ++ni) {
            const int col = s0 + ni * 16 + n;
            #pragma unroll
            for (int r = 0; r < 8; ++r) {
                const int row = t0 + mi * 16 + r + 8 * h;
                out[(size_t)row * SLEN + col] =
                    (col < len) ? acc[mi][ni][r] : -__builtin_inff();
            }
        }
}

// -------- Kernel 2: in-place row softmax over align ----------------------
__global__ __launch_bounds__(256) void softmax_kernel(float* __restrict__ align) {
    __shared__ float red[256];
    float* p = align + (size_t)blockIdx.x * SLEN;
    const int tid = threadIdx.x;
    float4 v = ((const float4*)p)[tid];
    float m = fmaxf(fmaxf(v.x, v.y), fmaxf(v.z, v.w));
    red[tid] = m;
    __syncthreads();
    for (int s = 128; s > 0; s >>= 1) {
        if (tid < s) red[tid] = fmaxf(red[tid], red[tid + s]);
        __syncthreads();
    }
    m = red[0];
    __syncthreads();
    float4 e;
    e.x = __expf(v.x - m); e.y = __expf(v.y - m);
    e.z = __expf(v.z - m); e.w = __expf(v.w - m);
    red[tid] = e.x + e.y + e.z + e.w;
    __syncthreads();
    for (int s = 128; s > 0; s >>= 1) {
        if (tid < s) red[tid] += red[tid + s];
        __syncthreads();
    }
    const float inv = 1.0f / red[0];
    e.x *= inv; e.y *= inv; e.z *= inv; e.w *= inv;
    ((float4*)p)[tid] = e;
}

// -------- Kernel 3: c = align @ src -> attn_h region (temporary) ---------
__global__ __launch_bounds__(256) void context_kernel(const float* __restrict__ align,
                                                      const float* __restrict__ src,
                                                      float* __restrict__ ctx) {
    const int lane = threadIdx.x & 31;
    int w = blockIdx.x * 8 + (threadIdx.x >> 5);
    const int nt = w & 7;   w >>= 3;   // 8 tiles of 64 along dim
    const int mt = w & 31;  w >>= 5;   // 32 tiles of 32 along tgt
    const int b  = w;
    const int t0 = mt * 32, n0 = nt * 64;
    const float* A = align + (size_t)b * TLEN * SLEN;  // [t, s]
    const float* B = src   + (size_t)b * SLEN * DIM;   // [s, d] row-major KxN

    v8f acc[2][4] = {};
    #pragma unroll 2
    for (int k0 = 0; k0 < SLEN; k0 += 4) {
        v2f a0 = frag_rk(A, t0,      SLEN, k0, lane);
        v2f a1 = frag_rk(A, t0 + 16, SLEN, k0, lane);
        v2f b0 = frag_kn(B, k0, n0,      DIM, lane);
        v2f b1 = frag_kn(B, k0, n0 + 16, DIM, lane);
        v2f b2 = frag_kn(B, k0, n0 + 32, DIM, lane);
        v2f b3 = frag_kn(B, k0, n0 + 48, DIM, lane);
        acc[0][0] = wmma4(a0, b0, acc[0][0]);
        acc[0][1] = wmma4(a0, b1, acc[0][1]);
        acc[0][2] = wmma4(a0, b2, acc[0][2]);
        acc[0][3] = wmma4(a0, b3, acc[0][3]);
        acc[1][0] = wmma4(a1, b0, acc[1][0]);
        acc[1][1] = wmma4(a1, b1, acc[1][1]);
        acc[1][2] = wmma4(a1, b2, acc[1][2]);
        acc[1][3] = wmma4(a1, b3, acc[1][3]);
    }
    float* out = ctx + (size_t)b * TLEN * DIM;
    const int n = lane & 15, h = lane >> 4;
    #pragma unroll
    for (int mi = 0; mi < 2; ++mi)
        #pragma unroll
        for (int ni = 0; ni < 4; ++ni)
            #pragma unroll
            for (int r = 0; r < 8; ++r)
                out[(size_t)(t0 + mi * 16 + r + 8 * h) * DIM + n0 + ni * 16 + n] =
                    acc[mi][ni][r];
}

// -------- Kernel 4: attn_h = c @ Wc^T + tgt @ Wt^T, in place -------------
// Block owns 16 tgt rows; c rows staged to padded LDS first, so the in-place
// overwrite of the attn_h region is race-free.
#define LDSPITCH 516   // 512 + 4 floats pad -> conflict-free b64 LDS reads
__global__ __launch_bounds__(256) void output_kernel(const float* __restrict__ tgt,
                                                     const float* __restrict__ wout,
                                                     float* __restrict__ attnh) {
    __shared__ float cs[16 * LDSPITCH];
    const int mt = blockIdx.x & 63;          // 64 tiles of 16 rows
    const int b  = blockIdx.x >> 6;
    const int t0 = mt * 16;
    float* crow = attnh + ((size_t)b * TLEN + t0) * DIM;
    for (int i = threadIdx.x; i < 16 * (DIM / 4); i += 256) {
        const int r = i >> 7, c4 = i & 127;
        const float4 v = ((const float4*)(crow + (size_t)r * DIM))[c4];
        *(float4*)(&cs[r * LDSPITCH + c4 * 4]) = v;
    }
    __syncthreads();

    const int lane = threadIdx.x & 31;
    const int n0 = (threadIdx.x >> 5) * 64;  // 8 waves cover 512 outputs
    const float* T = tgt + ((size_t)b * TLEN + t0) * DIM;
    const int m = lane & 15, h = lane >> 4;

    v8f acc[4] = {};
    #pragma unroll 4
    for (int k0 = 0; k0 < DIM; k0 += 4) {    // c part: A from LDS
        v2f a  = *(const v2f*)(&cs[m * LDSPITCH + k0 + 2 * h]);
        v2f b0 = frag_rk(wout, n0,      2 * DIM, k0, lane);
        v2f b1 = frag_rk(wout, n0 + 16, 2 * DIM, k0, lane);
        v2f b2 = frag_rk(wout, n0 + 32, 2 * DIM, k0, lane);
        v2f b3 = frag_rk(wout, n0 + 48, 2 * DIM, k0, lane);
        acc[0] = wmma4(a, b0, acc[0]);
        acc[1] = wmma4(a, b1, acc[1]);
        acc[2] = wmma4(a, b2, acc[2]);
        acc[3] = wmma4(a, b3, acc[3]);
    }
    #pragma unroll 4
    for (int k0 = 0; k0 < DIM; k0 += 4) {    // tgt part: w_out cols 512..1023
        v2f a  = frag_rk(T, 0, DIM, k0, lane);
        v2f b0 = frag_rk(wout, n0,      2 * DIM, DIM + k0, lane);
        v2f b1 = frag_rk(wout, n0 + 16, 2 * DIM, DIM + k0, lane);
        v2f b2 = frag_rk(wout, n0 + 32, 2 * DIM, DIM + k0, lane);
        v2f b3 = frag_rk(wout, n0 + 48, 2 * DIM, DIM + k0, lane);
        acc[0] = wmma4(a, b0, acc[0]);
        acc[1] = wmma4(a, b1, acc[1]);
        acc[2] = wmma4(a, b2, acc[2]);
        acc[3] = wmma4(a, b3, acc[3]);
    }
    const int n = lane & 15;
    #pragma unroll
    for (int ni = 0; ni < 4; ++ni)
        #pragma unroll
        for (int r = 0; r < 8; ++r)
            attnh[((size_t)b * TLEN + t0 + r + 8 * h) * DIM + n0 + ni * 16 + n] =
                acc[ni][r];
}

extern "C" void kernel_launch(void* const* d_in, const int* in_sizes, int n_in,
                              void* d_out, int out_size, void* d_ws, size_t ws_size,
                              hipStream_t stream) {
    const float* src  = (const float*)d_in[0];
    const float* tgt  = (const float*)d_in[1];
    const int*   lens = (const int*)d_in[2];   // JAX x64 off -> int32
    const float* wout = (const float*)d_in[3];

    float* attnh = (float*)d_out;
    float* align = attnh + (size_t)BZ * TLEN * DIM;

    // 1) masked scores -> align
    score_kernel<<<(BZ * 32 * 16) / 8, 256, 0, stream>>>(src, tgt, lens, align);
    // 2) softmax rows in place
    softmax_kernel<<<BZ * TLEN, 256, 0, stream>>>(align);
    // 3) context c -> attn_h region (temporary)
    context_kernel<<<(BZ * 32 * 8) / 8, 256, 0, stream>>>(align, src, attnh);
    // 4) attn_h = c@Wc^T + tgt@Wt^T, overwriting c in place
    output_kernel<<<BZ * (TLEN / 16), 256, 0, stream>>>(tgt, wout, attnh);
}